// SGC_58858231824466
// MI455X (gfx1250) — compile-verified
//
#include <hip/hip_runtime.h>

typedef float v2f __attribute__((ext_vector_type(2)));
typedef float v8f __attribute__((ext_vector_type(8)));

#define N_NODES 100000
#define N_EDGES 1600000
#define F_IN    128
#define F_OUT   64

// ---------------------------------------------------------------- degree ----
__global__ void sgc_deg_init(float* __restrict__ deg) {
  int i = blockIdx.x * blockDim.x + threadIdx.x;
  if (i < N_NODES) deg[i] = 1.0f;               // self-loop contributes 1
}

__global__ void sgc_deg_count(const long long* __restrict__ col,
                              float* __restrict__ deg) {
  int e = blockIdx.x * blockDim.x + threadIdx.x;
  if (e < N_EDGES) atomicAdd(&deg[(int)col[e]], 1.0f);
}

__global__ void sgc_deg_to_dinv(float* __restrict__ deg) {
  int i = blockIdx.x * blockDim.x + threadIdx.x;
  if (i < N_NODES) deg[i] = rsqrtf(deg[i]);     // deg >= 1 always (self-loop)
}

// ------------------------------------------------------------ propagation ---
// h_next[i] = dinv[i]^2 * h_cur[i]   (self-loop term, also zero-initializes)
__global__ void sgc_seed(const float* __restrict__ hcur,
                         const float* __restrict__ dinv,
                         float* __restrict__ hnext) {
  int t = blockIdx.x * blockDim.x + threadIdx.x;
  if (t < N_NODES * F_IN) {
    int i = t >> 7;                             // F_IN == 128
    float s = dinv[i];
    hnext[t] = s * s * hcur[t];
  }
}

// one wave (32 lanes) per edge; each lane moves one float4 (128 feats total)
__global__ void sgc_scatter(const long long* __restrict__ row,
                            const long long* __restrict__ col,
                            const float* __restrict__ dinv,
                            const float* __restrict__ hcur,
                            float* __restrict__ hnext) {
  long long t = (long long)blockIdx.x * blockDim.x + threadIdx.x;
  int e    = (int)(t >> 5);
  int lane = (int)(t & 31);
  if (e >= N_EDGES) return;
  int r = (int)row[e];
  int c = (int)col[e];
  float nrm = dinv[r] * dinv[c];
  const float4 v = *(const float4*)(hcur + (size_t)r * F_IN + lane * 4);
  float* dst = hnext + (size_t)c * F_IN + lane * 4;
  atomicAdd(dst + 0, nrm * v.x);
  atomicAdd(dst + 1, nrm * v.y);
  atomicAdd(dst + 2, nrm * v.z);
  atomicAdd(dst + 3, nrm * v.w);
}

// ------------------------------------------------------- dense layer (WMMA) -
// out[100000,64] = h[100000,128] @ W^T + b, W is [64,128] row-major.
// One wave -> one 16x16 tile. 4 waves/block cover N=64; grid.x covers M.
__global__ void __launch_bounds__(128) sgc_gemm_wmma(
    const float* __restrict__ h, const float* __restrict__ W,
    const float* __restrict__ b, float* __restrict__ out) {
  int lane = threadIdx.x & 31;
  int wv   = threadIdx.x >> 5;                  // n-tile 0..3
  int m0   = blockIdx.x * 16;                   // 100000 = 6250*16, no tail
  int n0   = wv * 16;
  int idx  = lane & 15;                         // A: row M, B: col N, C/D: col N
  int khalf = (lane >> 4) << 1;                 // lanes 0-15 -> K{0,1}; 16-31 -> K{2,3}

  float bv = b[n0 + idx];
  v8f c = {bv, bv, bv, bv, bv, bv, bv, bv};

  const float* hrow = h + (size_t)(m0 + idx) * F_IN;   // A[M=idx][k..]
  const float* wrow = W + (size_t)(n0 + idx) * F_IN;   // B[k][N=idx] = W[N=idx][k]

#pragma unroll
  for (int k = 0; k < F_IN; k += 4) {
    v2f a  = *(const v2f*)(hrow + k + khalf);
    v2f bb = *(const v2f*)(wrow + k + khalf);
    // D = A(16x4 f32) * B(4x16 f32) + C, exact fp32 path
    c = __builtin_amdgcn_wmma_f32_16x16x4_f32(
        /*neg_a=*/false, a, /*neg_b=*/false, bb,
        /*c_mod=*/(short)0, c, /*reuse_a=*/false, /*reuse_b=*/false);
  }

  int mBase = (lane >> 4) * 8;                  // lanes>=16 hold M=8..15
  float* o = out + (size_t)(m0 + mBase) * F_OUT + n0 + idx;
#pragma unroll
  for (int r = 0; r < 8; ++r) o[(size_t)r * F_OUT] = c[r];
}

// ---------------------------------------------------------------- launcher --
extern "C" void kernel_launch(void* const* d_in, const int* in_sizes, int n_in,
                              void* d_out, int out_size, void* d_ws, size_t ws_size,
                              hipStream_t stream) {
  const float*     x   = (const float*)d_in[0];       // [100000,128]
  const long long* ei  = (const long long*)d_in[1];   // [2,1600000] int64
  const float*     W   = (const float*)d_in[2];       // [64,128]
  const float*     b   = (const float*)d_in[3];       // [64]
  float*           out = (float*)d_out;               // [100000,64]

  const long long* row = ei;                 // source j
  const long long* col = ei + N_EDGES;       // target i

  float* deg = (float*)d_ws;                                 // N_NODES (-> dinv)
  float* hA  = deg + N_NODES;                                // N_NODES*128
  float* hB  = hA + (size_t)N_NODES * F_IN;                  // N_NODES*128

  const int B256 = 256;
  int gNode = (N_NODES + B256 - 1) / B256;
  int gEdge = (N_EDGES + B256 - 1) / B256;
  int gFeat = (N_NODES * F_IN + B256 - 1) / B256;
  int gScat = (int)(((long long)N_EDGES * 32 + B256 - 1) / B256);

  sgc_deg_init   <<<gNode, B256, 0, stream>>>(deg);
  sgc_deg_count  <<<gEdge, B256, 0, stream>>>(col, deg);
  sgc_deg_to_dinv<<<gNode, B256, 0, stream>>>(deg);          // deg now holds dinv

  // hop 1: x -> hA
  sgc_seed   <<<gFeat, B256, 0, stream>>>(x, deg, hA);
  sgc_scatter<<<gScat, B256, 0, stream>>>(row, col, deg, x, hA);
  // hop 2: hA -> hB
  sgc_seed   <<<gFeat, B256, 0, stream>>>(hA, deg, hB);
  sgc_scatter<<<gScat, B256, 0, stream>>>(row, col, deg, hA, hB);
  // hop 3: hB -> hA
  sgc_seed   <<<gFeat, B256, 0, stream>>>(hB, deg, hA);
  sgc_scatter<<<gScat, B256, 0, stream>>>(row, col, deg, hB, hA);

  // dense layer with fp32 WMMA
  sgc_gemm_wmma<<<N_NODES / 16, 128, 0, stream>>>(hA, W, b, out);
}